// Model_81698867904907
// MI455X (gfx1250) — compile-verified
//
#include <hip/hip_runtime.h>
#include <hip/hip_bf16.h>

// ---------------------------------------------------------------------------
// Stage 1: per-element VMD (thread-per-element, register-resident state).
// Stage 2: batched LSTM via V_WMMA_F32_16X16X4_F32 (16 batch rows / wave).
// ---------------------------------------------------------------------------

#define VMD_B      65536
#define VMD_ALPHA  2000.0f
#define VMD_TOL    1e-7f
#define VMD_EPS    2.2204460492503131e-16f
#define VMD_NITER  500

// cos(k*pi/6), sin(k*pi/6) for k=0..11 (exact table, folded at compile time)
#define SQ32 0.86602540378443864676f
__device__ __constant__ const float kC12[12] =
    {1.f, SQ32, 0.5f, 0.f, -0.5f, -SQ32, -1.f, -SQ32, -0.5f, 0.f, 0.5f, SQ32};
__device__ __constant__ const float kS12[12] =
    {0.f, 0.5f, SQ32, 1.f, SQ32, 0.5f, 0.f, -0.5f, -SQ32, -1.f, -SQ32, -0.5f};

__global__ __launch_bounds__(256) void vmd_kernel(const float* __restrict__ x,
                                                  float* __restrict__ u_out)
{
    const int b = blockIdx.x * 256 + threadIdx.x;
    const float* xb = x + (size_t)b * 720;

    // f6 = x_enc[:, [-96,-97,-98,-192,-193,-194]]  (SEQ_LEN=720)
    const float f0 = xb[624], f1 = xb[623], f2 = xb[622];
    const float f3 = xb[528], f4 = xb[527], f5 = xb[526];
    const float fM[12] = {f2,f1,f0,f0,f1,f2,f3,f4,f5,f5,f4,f3};

    // f_hat_plus[6+p] = F[p] = sum_n fM[n] * exp(-i*2*pi*p*n/12), p=0..5
    float fr[6], fi[6];
#pragma unroll
    for (int p = 0; p < 6; ++p) {
        float sr = 0.f, si = 0.f;
#pragma unroll
        for (int n = 0; n < 12; ++n) {
            const int idx = (p * n) % 12;
            sr += fM[n] * kC12[idx];
            si -= fM[n] * kS12[idx];
        }
        fr[p] = sr; fi[p] = si;
    }

    // VMD state: only positive-frequency half is ever nonzero (TAU=0, lam=0).
    float ur[3][6] = {}, ui[3][6] = {};
    float nur[3][6], nui[3][6];
    float skr[6] = {}, ski[6] = {};            // sum_uk (upper half)
    float om[3] = {0.0f, 1.0f/6.0f, 1.0f/3.0f};

    float ud = VMD_TOL + VMD_EPS;
    int it = 0;
    while (ud > VMD_TOL && it < VMD_NITER - 1) {
        float nom[3];
#pragma unroll
        for (int k = 0; k < 3; ++k) {
            float ps = 0.f, pw = 0.f;
            const float o = om[k];
#pragma unroll
            for (int j = 0; j < 6; ++j) {
                float pr, pi2;
                if (k == 0) { pr = ur[2][j] - ur[0][j];       pi2 = ui[2][j] - ui[0][j]; }
                else        { pr = nur[k-1][j] - ur[k][j];    pi2 = nui[k-1][j] - ui[k][j]; }
                skr[j] += pr; ski[j] += pi2;
                const float d   = (float)j * (1.0f/12.0f) - o;
                const float den = 1.0f + VMD_ALPHA * d * d;
                const float rr  = (fr[j] - skr[j]) / den;
                const float ii  = (fi[j] - ski[j]) / den;
                nur[k][j] = rr; nui[k][j] = ii;
                const float p = rr*rr + ii*ii;
                ps += p;
                pw += (float)j * (1.0f/12.0f) * p;
            }
            nom[k] = pw / ps;
        }
        float diff = 0.f;
#pragma unroll
        for (int k = 0; k < 3; ++k)
#pragma unroll
            for (int j = 0; j < 6; ++j) {
                const float dr = nur[k][j] - ur[k][j];
                const float di = nui[k][j] - ui[k][j];
                diff += dr*dr + di*di;
                ur[k][j] = nur[k][j]; ui[k][j] = nui[k][j];
            }
        om[0] = nom[0]; om[1] = nom[1]; om[2] = nom[2];
        ud = VMD_EPS + diff * (1.0f/12.0f);
        ++it;
    }

    // Reconstruction: u_t[t] = ((-1)^t/12) * Re( sum_m u_hat[m] e^{i*pi*t*m/6} ),
    // with u_hat[0]=conj(u11), u_hat[1..6]=conj(u11..u6), u_hat[7..11]=u7..u11.
    const int   jmap[12] = {5,5,4,3,2,1,0,1,2,3,4,5};
    const float sgn[12]  = {1,1,1,1,1,1,1,-1,-1,-1,-1,-1}; // conj:+b*s  plain:-b*s
    float* ob = u_out + (size_t)b * 18;
#pragma unroll
    for (int s = 0; s < 6; ++s) {
        const int t = s + 3;
        const float sc = ((t & 1) ? -1.0f : 1.0f) * (1.0f/12.0f);
#pragma unroll
        for (int k = 0; k < 3; ++k) {
            float acc = 0.f;
#pragma unroll
            for (int m = 0; m < 12; ++m) {
                const int idx = (t * m) % 12;
                const int j = jmap[m];
                acc += ur[k][j] * kC12[idx] + sgn[m] * ui[k][j] * kS12[idx];
            }
            ob[s*3 + k] = acc * sc;   // layout [b][t][mode]
        }
    }
}

// ---------------------------------------------------------------------------
// Stage 2: LSTM (in=3, hid=10, 6 steps) as WMMA f32 16x16x4 batched GEMM.
// gates(16x40) = [x|h](16x13, K padded to 16) @ Wcat^T ; 3 N-tiles x 4 K-chunks.
// ---------------------------------------------------------------------------

typedef __attribute__((ext_vector_type(2))) float v2f;
typedef __attribute__((ext_vector_type(8))) float v8f;

#define LSTM_WAVES 4

__device__ __forceinline__ float wcat(const float* __restrict__ Wih,
                                      const float* __restrict__ Whh,
                                      int n, int k)
{
    if (n >= 40 || k >= 13) return 0.0f;
    return (k < 3) ? Wih[n*3 + k] : Whh[n*10 + (k - 3)];
}

__global__ __launch_bounds__(LSTM_WAVES*32) void lstm_wmma_kernel(
    const float* __restrict__ u_all,
    const float* __restrict__ Wih, const float* __restrict__ Whh,
    const float* __restrict__ bih, const float* __restrict__ bhh,
    const float* __restrict__ Wlin, const float* __restrict__ blin,
    float* __restrict__ out)
{
    __shared__ float gates[LSTM_WAVES][16][48];
    __shared__ float hbuf[LSTM_WAVES][16][10];
    __shared__ float cbuf[LSTM_WAVES][16][10];

    const int lane = threadIdx.x & 31;
    const int wv   = threadIdx.x >> 5;
    const int m    = lane & 15;      // row within tile / column within tile
    const int hi   = lane >> 4;      // half-wave selector
    const long elem = (long)blockIdx.x * (LSTM_WAVES*16) + wv*16 + m;

    // B fragments: B[k][n] = Wcat[n][k]; VGPR0 holds rows {0,2}, VGPR1 rows {1,3}.
    v2f  bf[3][4];
    float bias[3];
#pragma unroll
    for (int nt = 0; nt < 3; ++nt) {
        const int n = nt*16 + m;
        bias[nt] = (n < 40) ? (bih[n] + bhh[n]) : 0.0f;
#pragma unroll
        for (int kc = 0; kc < 4; ++kc) {
            const int k0 = kc*4 + (hi ? 2 : 0);
            bf[nt][kc].x = wcat(Wih, Whh, n, k0);
            bf[nt][kc].y = wcat(Wih, Whh, n, k0 + 1);
        }
    }

    // zero h, c (160 items / wave, 5 per lane)
#pragma unroll
    for (int q = 0; q < 5; ++q) {
        const int idx = lane*5 + q;
        hbuf[wv][idx/10][idx%10] = 0.0f;
        cbuf[wv][idx/10][idx%10] = 0.0f;
    }
    __syncthreads();

    for (int t = 0; t < 6; ++t) {
        float xr[3];
#pragma unroll
        for (int k = 0; k < 3; ++k) xr[k] = u_all[elem*18 + t*3 + k];

        auto xh = [&](int k) -> float {
            if (k < 3)  return xr[k];
            if (k < 13) return hbuf[wv][m][k - 3];
            return 0.0f;
        };

        // A fragments: 16x4 f32, VGPR0 = K{0,2}, VGPR1 = K{1,3}.
        v2f af[4];
#pragma unroll
        for (int kc = 0; kc < 4; ++kc) {
            const int k0 = kc*4 + (hi ? 2 : 0);
            af[kc].x = xh(k0);
            af[kc].y = xh(k0 + 1);
        }

        v8f acc[3];
#pragma unroll
        for (int nt = 0; nt < 3; ++nt)
#pragma unroll
            for (int r = 0; r < 8; ++r) acc[nt][r] = bias[nt];

#pragma unroll
        for (int nt = 0; nt < 3; ++nt)
#pragma unroll
            for (int kc = 0; kc < 4; ++kc)
                acc[nt] = __builtin_amdgcn_wmma_f32_16x16x4_f32(
                    false, af[kc], false, bf[nt][kc],
                    (short)0, acc[nt], false, false);

        // D layout: VGPR r -> row (r + 8*hi), column = lane%16 within N-tile.
#pragma unroll
        for (int nt = 0; nt < 3; ++nt)
#pragma unroll
            for (int r = 0; r < 8; ++r)
                gates[wv][r + 8*hi][nt*16 + m] = acc[nt][r];
        __syncthreads();

        // LSTM nonlinearity: 5 (row, hidden) items per lane.
#pragma unroll
        for (int q = 0; q < 5; ++q) {
            const int idx = lane*5 + q;
            const int mm = idx / 10, hh = idx % 10;
            const float gi = gates[wv][mm][hh];
            const float gf = gates[wv][mm][10 + hh];
            const float gg = gates[wv][mm][20 + hh];
            const float go = gates[wv][mm][30 + hh];
            const float si = 1.0f / (1.0f + expf(-gi));
            const float sf = 1.0f / (1.0f + expf(-gf));
            const float so = 1.0f / (1.0f + expf(-go));
            const float cn = sf * cbuf[wv][mm][hh] + si * tanhf(gg);
            cbuf[wv][mm][hh] = cn;
            hbuf[wv][mm][hh] = so * tanhf(cn);
        }
        __syncthreads();
    }

    // final linear head on h(t=5)
    if (lane < 16) {
        float d = 0.0f;
#pragma unroll
        for (int h = 0; h < 10; ++h) d += hbuf[wv][m][h] * Wlin[h];
        out[elem] = d + blin[0];
    }
}

// ---------------------------------------------------------------------------

extern "C" void kernel_launch(void* const* d_in, const int* in_sizes, int n_in,
                              void* d_out, int out_size, void* d_ws, size_t ws_size,
                              hipStream_t stream)
{
    (void)in_sizes; (void)n_in; (void)out_size; (void)ws_size;
    const float* x_enc = (const float*)d_in[0];
    const float* Wih   = (const float*)d_in[4];
    const float* Whh   = (const float*)d_in[5];
    const float* bih   = (const float*)d_in[6];
    const float* bhh   = (const float*)d_in[7];
    const float* Wlin  = (const float*)d_in[8];
    const float* blin  = (const float*)d_in[9];
    float* out   = (float*)d_out;
    float* u_all = (float*)d_ws;     // B*6*3 floats

    vmd_kernel<<<VMD_B/256, 256, 0, stream>>>(x_enc, u_all);
    lstm_wmma_kernel<<<VMD_B/(LSTM_WAVES*16), LSTM_WAVES*32, 0, stream>>>(
        u_all, Wih, Whh, bih, bhh, Wlin, blin, out);
}